// LightGCNModel_7868380086591
// MI455X (gfx1250) — compile-verified
//
#include <hip/hip_runtime.h>

// ---------------- problem constants (from reference) ----------------
#define NUM_USERS_    400000
#define NUM_PRODUCTS_ 300000
#define NUM_NODES_    700000          // users + products
#define EMBED_DIM_    64
#define NUM_EDGES_    1200000
#define VEC_PER_NODE  16              // 64 floats = 16 float4
#define TOTAL_VEC     (NUM_NODES_ * VEC_PER_NODE)   // 11.2M float4

typedef float __attribute__((ext_vector_type(4))) f4;
typedef int v4i_vs __attribute__((vector_size(16)));   // matches builtin's V4i

// ---------------- CDNA5 async global->LDS path (gfx1250) ----------------
#if defined(__HIP_DEVICE_COMPILE__) && defined(__gfx1250__) &&                 \
    __has_builtin(__builtin_amdgcn_global_load_async_to_lds_b128) &&           \
    __has_builtin(__builtin_amdgcn_s_wait_asynccnt)
#define HAVE_ASYNC_LDS 1
#else
#define HAVE_ASYNC_LDS 0
#endif

#if HAVE_ASYNC_LDS
// signature (probe-confirmed arity): (v4i AS1* gsrc, v4i AS3* ldst, imm off, imm cpol)
#define ASYNC_COPY16(gsrc, ldst)                                               \
  __builtin_amdgcn_global_load_async_to_lds_b128(                              \
      (__attribute__((address_space(1))) v4i_vs*)(unsigned long long)(gsrc),   \
      (__attribute__((address_space(3))) v4i_vs*)(unsigned)(unsigned long long)(ldst), \
      0, 0)
#define ASYNC_WAIT(n) __builtin_amdgcn_s_wait_asynccnt(n)
#endif

// ---------------- small setup kernels ----------------
__global__ void zero_f32_kernel(float* __restrict__ p, int n) {
  int i = blockIdx.x * blockDim.x + threadIdx.x;
  if (i < n) p[i] = 0.0f;
}

// deg[row[e]] += 1  (row = edge_index[0], per reference)
__global__ void degree_kernel(const int* __restrict__ ei, float* __restrict__ deg) {
  int e = blockIdx.x * blockDim.x + threadIdx.x;
  if (e < NUM_EDGES_) unsafeAtomicAdd(&deg[ei[e]], 1.0f);
}

// deg -> deg^{-1/2}  (max(deg,1) first); accurate 1/sqrtf, once per node
__global__ void dinv_kernel(float* __restrict__ deg) {
  int i = blockIdx.x * blockDim.x + threadIdx.x;
  if (i < NUM_NODES_) {
    float d = fmaxf(deg[i], 1.0f);
    deg[i] = 1.0f / sqrtf(d);
  }
}

// acc(d_out) = embA = concat(user_emb, product_emb); embB = 0
__global__ void init_kernel(const f4* __restrict__ ue, const f4* __restrict__ pe,
                            f4* __restrict__ acc, f4* __restrict__ eA,
                            f4* __restrict__ eB) {
  int i = blockIdx.x * blockDim.x + threadIdx.x;
  if (i >= TOTAL_VEC) return;
  const int uv = NUM_USERS_ * VEC_PER_NODE;
  // NT loads: inputs are streamed once, keep L2 for the embedding tables
  f4 v = (i < uv) ? __builtin_nontemporal_load(&ue[i])
                  : __builtin_nontemporal_load(&pe[i - uv]);
  acc[i] = v;
  eA[i]  = v;
  f4 z = {0.f, 0.f, 0.f, 0.f};
  eB[i] = z;
}

// ---------------- edge-parallel propagate: embnext[col] += emb[row]*norm ----
// 16 lanes per edge (each lane: one float4 chunk of the 64-float row).
// Lane 0 / lane 16 of each wave load indices + norm, broadcast via shfl.
// Gathers are software-pipelined (depth 2) through LDS with the CDNA5
// async global->LDS instructions tracked by ASYNCcnt.
#define SCATTER_GRID 4096
#define EPB 16   // edges per block per iteration (256 threads / 16 lanes)

__global__ __launch_bounds__(256) void scatter_kernel(
    const int* __restrict__ ei, const float* __restrict__ dinv,
    const float* __restrict__ embcur, float* __restrict__ embnext) {
#if HAVE_ASYNC_LDS
  __shared__ f4 stage[2][256];   // 8 KB double buffer
#endif
  const int tid  = threadIdx.x;
  const int lane = tid & 31;
  const int bsrc = lane & 16;    // broadcast source lane (0 or 16) in wave
  const int sub  = tid >> 4;     // edge slot 0..15 within block
  const int ch   = tid & 15;     // float4 chunk 0..15 within row
  const long stride = (long)gridDim.x * EPB;
  long b = (long)blockIdx.x * EPB;   // always < NUM_EDGES_ for grid=4096

  // ---- prologue: indices/norm (+ async gather) for stage 0 ----
  int rowC, colC; float normC;
  {
    long e = b + sub; if (e >= NUM_EDGES_) e = NUM_EDGES_ - 1;
    int r = 0, c = 0; float nm = 0.f;
    if ((lane & 15) == 0) {
      r  = ei[e];
      c  = ei[NUM_EDGES_ + e];
      nm = dinv[r] * dinv[c];
    }
    rowC  = __shfl(r, bsrc, 32);
    colC  = __shfl(c, bsrc, 32);
    normC = __shfl(nm, bsrc, 32);
#if HAVE_ASYNC_LDS
    ASYNC_COPY16(embcur + ((long)rowC * EMBED_DIM_ + ch * 4), &stage[0][tid]);
#else
    __builtin_prefetch(embcur + ((long)rowC * EMBED_DIM_ + ch * 4), 0, 3);
#endif
  }

  int s = 0; (void)s;
  for (; b < NUM_EDGES_; b += stride) {
    const long bn = b + stride;
    const bool has_next = bn < NUM_EDGES_;   // uniform across block
    int rowN = 0, colN = 0; float normN = 0.f;
    if (has_next) {
      long en = bn + sub; if (en >= NUM_EDGES_) en = NUM_EDGES_ - 1;
      int r = 0, c = 0; float nm = 0.f;
      if ((lane & 15) == 0) {
        r  = ei[en];
        c  = ei[NUM_EDGES_ + en];
        nm = dinv[r] * dinv[c];
      }
      rowN  = __shfl(r, bsrc, 32);
      colN  = __shfl(c, bsrc, 32);
      normN = __shfl(nm, bsrc, 32);
#if HAVE_ASYNC_LDS
      ASYNC_COPY16(embcur + ((long)rowN * EMBED_DIM_ + ch * 4), &stage[s ^ 1][tid]);
#else
      __builtin_prefetch(embcur + ((long)rowN * EMBED_DIM_ + ch * 4), 0, 3);
#endif
    }

#if HAVE_ASYNC_LDS
    if (has_next) ASYNC_WAIT(1); else ASYNC_WAIT(0);
    asm volatile("" ::: "memory");          // keep ds_read below the wait
    f4 v = stage[s][tid];
#else
    f4 v = *(const f4*)(embcur + ((long)rowC * EMBED_DIM_ + ch * 4));
#endif

    long e = b + sub;
    if (e < NUM_EDGES_) {
      float* dst = embnext + ((long)colC * EMBED_DIM_ + ch * 4);
      unsafeAtomicAdd(dst + 0, v.x * normC);
      unsafeAtomicAdd(dst + 1, v.y * normC);
      unsafeAtomicAdd(dst + 2, v.z * normC);
      unsafeAtomicAdd(dst + 3, v.w * normC);
    }
    rowC = rowN; colC = colN; normC = normN;
#if HAVE_ASYNC_LDS
    s ^= 1;
#endif
  }
}

// acc += embnext ; re-zero the other ping-pong buffer ; last layer: *0.25 (NT)
__global__ void combine_kernel(const f4* __restrict__ nxt, f4* __restrict__ acc,
                               f4* __restrict__ zbuf, int last) {
  int i = blockIdx.x * blockDim.x + threadIdx.x;
  if (i >= TOTAL_VEC) return;
  f4 v = nxt[i];
  f4 a = acc[i];
  a.x += v.x; a.y += v.y; a.z += v.z; a.w += v.w;
  if (last) {
    a.x *= 0.25f; a.y *= 0.25f; a.z *= 0.25f; a.w *= 0.25f;
    __builtin_nontemporal_store(a, &acc[i]);   // final output: stream out
  } else {
    acc[i] = a;
    f4 z = {0.f, 0.f, 0.f, 0.f};
    zbuf[i] = z;                               // next scatter target
  }
}

// ---------------- launcher ----------------
extern "C" void kernel_launch(void* const* d_in, const int* in_sizes, int n_in,
                              void* d_out, int out_size, void* d_ws, size_t ws_size,
                              hipStream_t stream) {
  (void)in_sizes; (void)n_in; (void)out_size; (void)ws_size;
  const int*   ei   = (const int*)d_in[0];    // edge_index [2, E] flattened
  const float* uemb = (const float*)d_in[1];  // [NUM_USERS, 64]
  const float* pemb = (const float*)d_in[2];  // [NUM_PRODUCTS, 64]
  float* out = (float*)d_out;                 // [NUM_NODES, 64]

  // workspace: deg/dinv | embA | embB
  float* deg = (float*)d_ws;
  size_t off = ((size_t)NUM_NODES_ * sizeof(float) + 255) & ~(size_t)255;
  float* embA = (float*)((char*)d_ws + off);
  float* embB = embA + (size_t)NUM_NODES_ * EMBED_DIM_;

  const dim3 blk(256);
  const int gN = (NUM_NODES_ + 255) / 256;
  const int gE = (NUM_EDGES_ + 255) / 256;
  const int gV = (TOTAL_VEC + 255) / 256;

  zero_f32_kernel<<<gN, blk, 0, stream>>>(deg, NUM_NODES_);
  degree_kernel  <<<gE, blk, 0, stream>>>(ei, deg);
  dinv_kernel    <<<gN, blk, 0, stream>>>(deg);
  init_kernel    <<<gV, blk, 0, stream>>>((const f4*)uemb, (const f4*)pemb,
                                          (f4*)out, (f4*)embA, (f4*)embB);

  // layer 1: A -> B, acc += B, zero A
  scatter_kernel <<<SCATTER_GRID, blk, 0, stream>>>(ei, deg, embA, embB);
  combine_kernel <<<gV, blk, 0, stream>>>((const f4*)embB, (f4*)out, (f4*)embA, 0);
  // layer 2: B -> A, acc += A, zero B
  scatter_kernel <<<SCATTER_GRID, blk, 0, stream>>>(ei, deg, embB, embA);
  combine_kernel <<<gV, blk, 0, stream>>>((const f4*)embA, (f4*)out, (f4*)embB, 0);
  // layer 3: A -> B, out = (acc + B) / 4
  scatter_kernel <<<SCATTER_GRID, blk, 0, stream>>>(ei, deg, embA, embB);
  combine_kernel <<<gV, blk, 0, stream>>>((const f4*)embB, (f4*)out, (f4*)nullptr, 1);
}